// RecursiveNonlinearEquationSolver_44083544326153
// MI455X (gfx1250) — compile-verified
//
#include <hip/hip_runtime.h>

// Problem constants (match reference).
#define N_TIME   64
#define N_BATCH  8192
#define N_STATE  32
#define N_FORCE  16
#define DT_STEP  0.05f
// Picard contraction rate ~= DT*||W||_2 ~= 0.1  ->  12 iters => 1e-12, below fp32 eps.
#define PICARD_ITERS 12

typedef float v2f __attribute__((ext_vector_type(2)));
typedef float v4f __attribute__((ext_vector_type(4)));
typedef float v8f __attribute__((ext_vector_type(8)));

// Swap lane halves (lane <-> lane^16) via DS_SWIZZLE_B32, SWAPX16: xor=0x10,or=0,and=0x1f.
__device__ __forceinline__ float swap16(float x) {
  int r = __builtin_amdgcn_ds_swizzle(__builtin_bit_cast(int, x), 0x401f);
  return __builtin_bit_cast(float, r);
}

__device__ __forceinline__ float fast_tanh(float x) {
#if __has_builtin(__builtin_amdgcn_tanhf)
  return __builtin_amdgcn_tanhf(x);
#else
  // tanh(x) = (e-1)/(e+1), e = exp(2x) = exp2(2x*log2(e)); clamp avoids inf/inf.
  float xc = fminf(fmaxf(x, -15.0f), 15.0f);
  float e  = __builtin_amdgcn_exp2f(xc * 2.885390081777927f); // 2*log2(e)
  return (e - 1.0f) * __builtin_amdgcn_rcpf(e + 1.0f);
#endif
}

// D = A(16x4 f32) x B(4x16 f32) + C(16x16 f32), wave32 WMMA.
__device__ __forceinline__ v8f wmma_f32_16x16x4(v2f a, v2f b, v8f c) {
  return __builtin_amdgcn_wmma_f32_16x16x4_f32(false, a, false, b, (short)0, c,
                                               false, false);
}

// Per wave: 16 batch rows, marched through all 64 time steps.
// State kept in WMMA C/D layout of z.T = W @ y.T:
//   tile T (states 16T..16T+15), vgpr i, lane l  holds
//   y[batch = base + (l&15)][state = 16T + 8*(l>>4) + i]
__global__ void __launch_bounds__(256)
solver_kernel(const float* __restrict__ y0, const float* __restrict__ forces,
              const float* __restrict__ W,  const float* __restrict__ U,
              const float* __restrict__ b,  float* __restrict__ out) {
  const int lane = threadIdx.x & 31;
  const int wave = (int)((blockIdx.x * blockDim.x + threadIdx.x) >> 5);
  const int base = wave * 16;      // first batch row of this wave's tile
  const int ln   = lane & 15;      // batch index within tile (also A-row / B-col lane)
  const int hi   = lane >> 4;      // lane half
  const bool lanehi = (hi != 0);

  // ---- W as A-fragments: Mblock i (out states 16i..), Kblock j (in states 4j..).
  // A layout: lane m<16 -> {K=4j, 4j+1}; lane m+16 -> {K=4j+2, 4j+3}.
  v2f Wf[2][8];
#pragma unroll
  for (int i = 0; i < 2; ++i)
#pragma unroll
    for (int j = 0; j < 8; ++j)
      Wf[i][j] = *(const v2f*)(W + (16 * i + ln) * N_STATE + 4 * j + 2 * hi);

  // ---- U as A-fragments (K = 16 forces -> 4 K-blocks).
  v2f Uf[2][4];
#pragma unroll
  for (int i = 0; i < 2; ++i)
#pragma unroll
    for (int j = 0; j < 4; ++j)
      Uf[i][j] = *(const v2f*)(U + (16 * i + ln) * N_FORCE + 4 * j + 2 * hi);

  // ---- bias in D layout (used as WMMA C seed for fu).
  v8f bias[2];
#pragma unroll
  for (int T = 0; T < 2; ++T) {
    const float* pb = b + 16 * T + 8 * hi;
#pragma unroll
    for (int i = 0; i < 8; ++i) bias[T][i] = pb[i];
  }

  // ---- y_prev = y0 in D layout; also emit out[0] = y0.
  v8f yprev[2];
#pragma unroll
  for (int T = 0; T < 2; ++T) {
    const float* p = y0 + (size_t)(base + ln) * N_STATE + 16 * T + 8 * hi;
    v4f a0 = *(const v4f*)p;
    v4f a1 = *(const v4f*)(p + 4);
#pragma unroll
    for (int i = 0; i < 4; ++i) { yprev[T][i] = a0[i]; yprev[T][i + 4] = a1[i]; }
    float* q = out + (size_t)(base + ln) * N_STATE + 16 * T + 8 * hi;
    *(v4f*)q       = a0;
    *(v4f*)(q + 4) = a1;
  }

#pragma unroll 1
  for (int t = 1; t < N_TIME; ++t) {
    // ---- u_t as B-fragments: vgpr p, lane l -> forces[t][base+ln][4j + 2*hi + p].
    const float* fptr = forces + ((size_t)t * N_BATCH + base + ln) * N_FORCE;
    v2f uf[4];
#pragma unroll
    for (int j = 0; j < 4; ++j) uf[j] = *(const v2f*)(fptr + 4 * j + 2 * hi);

    // ---- fu.T = U @ u.T + b  (D layout), via WMMA with C seeded by bias.
    v8f fu[2];
#pragma unroll
    for (int T = 0; T < 2; ++T) {
      v8f acc = bias[T];
#pragma unroll
      for (int j = 0; j < 4; ++j) acc = wmma_f32_16x16x4(Uf[T][j], uf[j], acc);
      fu[T] = acc;
    }

    // ---- Picard iteration 0 (zero initial guess => z = fu).
    v8f y[2];
#pragma unroll
    for (int T = 0; T < 2; ++T)
#pragma unroll
      for (int i = 0; i < 8; ++i)
        y[T][i] = yprev[T][i] + DT_STEP * fast_tanh(fu[T][i]);

    // ---- Picard iterations with the W matvec.
#pragma unroll 1
    for (int it = 1; it < PICARD_ITERS; ++it) {
      // Rebuild B-fragments of y.T from the D-layout state: state s lives in
      // tile s>>4, vgpr s&7, lane-half (s>>3)&1; fragment (j,p) wants state
      // 4j+p in lanes<16 and 4j+2+p in lanes>=16 (same source half for both).
      v2f bf[8];
#pragma unroll
      for (int j = 0; j < 8; ++j) {
#pragma unroll
        for (int p = 0; p < 2; ++p) {
          const int s_lo = 4 * j + p;          // compile-time constants
          const int Ts   = s_lo >> 4;
          const int i_lo = s_lo & 7;
          const int i_hi = i_lo + 2;
          const int h    = (s_lo >> 3) & 1;    // half where source data lives
          float flo = y[Ts][i_lo];
          float fhi = y[Ts][i_hi];
          float sw   = swap16((h == 0) ? fhi : flo);  // unconditional, convergent-safe
          float keep = (h == 0) ? flo : fhi;
          bool take_sw = (h == 0) ? lanehi : !lanehi;
          bf[j][p] = take_sw ? sw : keep;
        }
      }
      // z.T = W @ y.T + fu ; two accumulator chains per tile to halve WMMA depth.
#pragma unroll
      for (int T = 0; T < 2; ++T) {
        v8f accA = fu[T];
        v8f accB = {};
#pragma unroll
        for (int j = 0; j < 4; ++j) accA = wmma_f32_16x16x4(Wf[T][j], bf[j], accA);
#pragma unroll
        for (int j = 4; j < 8; ++j) accB = wmma_f32_16x16x4(Wf[T][j], bf[j], accB);
        v8f z = accA + accB;
#pragma unroll
        for (int i = 0; i < 8; ++i)
          y[T][i] = yprev[T][i] + DT_STEP * fast_tanh(z[i]);
      }
    }

    // ---- store y_t, advance recursion.
#pragma unroll
    for (int T = 0; T < 2; ++T) {
      float* q = out + ((size_t)t * N_BATCH + base + ln) * N_STATE + 16 * T + 8 * hi;
      v4f s0, s1;
#pragma unroll
      for (int i = 0; i < 4; ++i) { s0[i] = y[T][i]; s1[i] = y[T][i + 4]; }
      *(v4f*)q       = s0;
      *(v4f*)(q + 4) = s1;
      yprev[T] = y[T];
    }
  }
}

extern "C" void kernel_launch(void* const* d_in, const int* in_sizes, int n_in,
                              void* d_out, int out_size, void* d_ws, size_t ws_size,
                              hipStream_t stream) {
  const float* y0     = (const float*)d_in[0];
  const float* forces = (const float*)d_in[1];
  const float* W      = (const float*)d_in[2];
  const float* U      = (const float*)d_in[3];
  const float* b      = (const float*)d_in[4];
  float* out          = (float*)d_out;

  // 8192 batch rows / 16 per wave = 512 waves; 8 waves (256 thr) per block -> 64 blocks.
  solver_kernel<<<dim3(N_BATCH / 16 / 8), dim3(256), 0, stream>>>(
      y0, forces, W, U, b, out);
}